// _AttentionGIM_36541581754920
// MI455X (gfx1250) — compile-verified
//
#include <hip/hip_runtime.h>

// ---------------------------------------------------------------------------
// Fused masked attention forward for MI455X (gfx1250, wave32, WMMA f32).
//   out [96,1024,64] and p_attn [96,1024,1024], fp32.
// One block = 64 query rows of one (b,h). Full 64x1028-padded score strip in
// LDS (~303 KB of the 320 KB/WGP), softmax in-place, p_attn streamed to HBM
// with GLOBAL_STORE_ASYNC_FROM_LDS_B128 overlapped under the p@V WMMA phase.
// ---------------------------------------------------------------------------

typedef __attribute__((ext_vector_type(2))) float v2f;
typedef __attribute__((ext_vector_type(8))) float v8f;
typedef __attribute__((ext_vector_type(4))) int   v4i;   // b128 payload type

constexpr int S  = 1024;
constexpr int D  = 64;
constexpr int H  = 12;
constexpr int BH = 96;          // B*H
constexpr int BM = 64;          // query rows per block
constexpr int BN = 64;          // key/value rows per tile

constexpr int QPITCH = 68;      // 64 + 4 pad: banks (4*row + col) % 64 distinct
constexpr int SPITCH = 1028;    // 1024 + 4 pad: same bank-conflict-free property

// LDS partition (in floats)
constexpr int OFF_Q    = 0;
constexpr int OFF_KV   = OFF_Q  + BM * QPITCH;   // K tile, reused for V tiles
constexpr int OFF_S    = OFF_KV + BN * QPITCH;   // 64 x 1028 score/prob strip
constexpr int OFF_MASK = OFF_S  + BM * SPITCH;   // 1024 mask floats (0/1)
constexpr int OFF_RED  = OFF_MASK + S;           // 256 reduction slots
constexpr int SMEM_FLOATS = OFF_RED + 256;       // 75,776 floats = 303,104 B

// -------- CDNA5 async global<->LDS path (guarded; falls back to sync) ------
#if defined(__AMDGCN__) && __has_builtin(__builtin_amdgcn_global_load_async_to_lds_b128) && \
    __has_builtin(__builtin_amdgcn_global_store_async_from_lds_b128)
#define HAVE_ASYNC_LDS 1
#else
#define HAVE_ASYNC_LDS 0
#endif

__device__ __forceinline__ void wait_async_zero() {
#if defined(__AMDGCN__)
#if __has_builtin(__builtin_amdgcn_s_wait_asynccnt)
    __builtin_amdgcn_s_wait_asynccnt(0);
#else
    asm volatile("s_wait_asynccnt 0x0" ::: "memory");
#endif
#endif
}

__global__ __launch_bounds__(256)
void attn_gim_fwd(const float* __restrict__ Qg,
                  const float* __restrict__ Kg,
                  const float* __restrict__ Vg,
                  const int*   __restrict__ Mg,
                  float*       __restrict__ Og,
                  float*       __restrict__ Pg) {
    extern __shared__ float smem[];
    float* sQ    = smem + OFF_Q;
    float* sKV   = smem + OFF_KV;
    float* sS    = smem + OFF_S;
    float* sMask = smem + OFF_MASK;
    float* sRed  = smem + OFF_RED;

    const int tid  = threadIdx.x;
    const int bh   = blockIdx.y;            // 0..95
    const int m0   = blockIdx.x * BM;       // query-row base
    const int b    = bh / H;

    const int w     = tid >> 5;             // wave id 0..7
    const int lane  = tid & 31;
    const int msub  = w & 3;                // 16-row sub-strip within the 64 rows
    const int np    = w >> 2;               // 0/1 -> owns n-subtiles {2np, 2np+1}
    const int l16   = lane & 15;
    const int hi    = lane >> 4;            // half-wave select
    const int khalf = hi << 1;              // K offset 0 / 2 (f32 A/B frag layout)
    const int hi8   = hi << 3;              // C frag: rows r / r+8

    // ---- stage Q tile + mask --------------------------------------------
    for (int i = tid; i < BM * (D / 4); i += 256) {
        const int row = i >> 4, c4 = (i & 15) << 2;
        *(float4*)&sQ[row * QPITCH + c4] =
            *(const float4*)&Qg[((size_t)bh * S + m0 + row) * D + c4];
    }
    for (int i = tid; i < S; i += 256)
        sMask[i] = Mg[b * S + i] ? 1.0f : 0.0f;

    // ---- phase 1: scores = mask ? (Q K^T)/8 : -1e9  into LDS ------------
    for (int nt = 0; nt < S / BN; ++nt) {
        __syncthreads();                         // sKV free from previous tile
        const int n0 = nt * BN;
#if HAVE_ASYNC_LDS
        for (int i = tid; i < BN * (D / 4); i += 256) {
            const int row = i >> 4, c4 = (i & 15) << 2;
            __builtin_amdgcn_global_load_async_to_lds_b128(
                (v4i*)&Kg[((size_t)bh * S + n0 + row) * D + c4],
                (v4i*)&sKV[row * QPITCH + c4], 0, 0);
        }
        wait_async_zero();
#else
        for (int i = tid; i < BN * (D / 4); i += 256) {
            const int row = i >> 4, c4 = (i & 15) << 2;
            *(float4*)&sKV[row * QPITCH + c4] =
                *(const float4*)&Kg[((size_t)bh * S + n0 + row) * D + c4];
        }
#endif
        if (nt + 1 < S / BN)
            __builtin_prefetch(&Kg[((size_t)bh * S + n0 + BN) * D + tid * 16], 0, 1);
        __syncthreads();

        v8f acc0 = {}; v8f acc1 = {};
        const float* aRow  = &sQ [(msub * 16 + l16) * QPITCH + khalf];
        const float* bRow0 = &sKV[((np * 2 + 0) * 16 + l16) * QPITCH + khalf];
        const float* bRow1 = &sKV[((np * 2 + 1) * 16 + l16) * QPITCH + khalf];
#pragma unroll
        for (int d0 = 0; d0 < D; d0 += 4) {
            v2f a  = *(const v2f*)(aRow  + d0);
            v2f b0 = *(const v2f*)(bRow0 + d0);
            v2f b1 = *(const v2f*)(bRow1 + d0);
            acc0 = __builtin_amdgcn_wmma_f32_16x16x4_f32(false, a, false, b0,
                                                         (short)0, acc0, false, false);
            acc1 = __builtin_amdgcn_wmma_f32_16x16x4_f32(false, a, false, b1,
                                                         (short)0, acc1, false, false);
        }
        const int col0 = n0 + (np * 2 + 0) * 16 + l16;
        const int col1 = col0 + 16;
        const float mk0 = sMask[col0], mk1 = sMask[col1];
#pragma unroll
        for (int r = 0; r < 8; ++r) {
            const int m = msub * 16 + r + hi8;
            sS[m * SPITCH + col0] = (mk0 != 0.f) ? acc0[r] * 0.125f : -1e9f;
            sS[m * SPITCH + col1] = (mk1 != 0.f) ? acc1[r] * 0.125f : -1e9f;
        }
    }
    __syncthreads();

    // ---- phase 2: row softmax in LDS, mask-zero, stream p to global -----
    {
        const int row  = tid >> 2;          // 4 threads per row
        const int part = tid & 3;
        float* rowp = &sS[row * SPITCH + part * 256];

        float lmax = -3.0e38f;
        for (int c = 0; c < 256; ++c) lmax = fmaxf(lmax, rowp[c]);
        sRed[tid] = lmax;
        __syncthreads();
        const float rmax = fmaxf(fmaxf(sRed[row * 4 + 0], sRed[row * 4 + 1]),
                                 fmaxf(sRed[row * 4 + 2], sRed[row * 4 + 3]));
        __syncthreads();

        float lsum = 0.f;
        for (int c = 0; c < 256; ++c) {
            const float e = expf(rowp[c] - rmax);
            rowp[c] = e;
            lsum += e;
        }
        sRed[tid] = lsum;
        __syncthreads();
        const float rinv = 1.0f / (sRed[row * 4 + 0] + sRed[row * 4 + 1] +
                                   sRed[row * 4 + 2] + sRed[row * 4 + 3]);
        const float* mrow = &sMask[part * 256];
        for (int c = 0; c < 256; ++c) rowp[c] = rowp[c] * rinv * mrow[c];
        __syncthreads();

#if HAVE_ASYNC_LDS
        // Fire-and-forget: LDS -> HBM direct, drains under the p@V phase.
        // Phase 3 never writes sS, so no wait is needed until kernel end.
        for (int i = tid; i < BM * (S / 4); i += 256) {
            const int r2 = i >> 8, c4 = (i & 255) << 2;
            __builtin_amdgcn_global_store_async_from_lds_b128(
                (v4i*)&Pg[((size_t)bh * S + m0 + r2) * S + c4],
                (v4i*)&sS[r2 * SPITCH + c4], 0, 0);
        }
#else
        for (int i = tid; i < BM * (S / 4); i += 256) {      // coalesced float4
            const int r2 = i >> 8, c4 = (i & 255) << 2;
            *(float4*)&Pg[((size_t)bh * S + m0 + r2) * S + c4] =
                *(const float4*)&sS[r2 * SPITCH + c4];
        }
#endif
    }

    // ---- phase 3: out = p @ V (V tiles on LOADcnt path, NOT async, so ----
    // ---- waiting for them never drains the in-flight p_attn stores) -----
    v8f o0 = {}; v8f o1 = {};
    const int nA = (np * 2 + 0) * 16 + l16;
    const int nB = nA + 16;
    for (int kt = 0; kt < S / BN; ++kt) {
        __syncthreads();                        // sKV free
        const int k0 = kt * BN;
        for (int i = tid; i < BN * (D / 4); i += 256) {
            const int row = i >> 4, c4 = (i & 15) << 2;
            *(float4*)&sKV[row * QPITCH + c4] =
                *(const float4*)&Vg[((size_t)bh * S + k0 + row) * D + c4];
        }
        if (kt + 1 < S / BN)
            __builtin_prefetch(&Vg[((size_t)bh * S + k0 + BN) * D + tid * 16], 0, 1);
        __syncthreads();

        const float* aRow = &sS[(msub * 16 + l16) * SPITCH + k0 + khalf];
#pragma unroll
        for (int kk = 0; kk < BN; kk += 4) {
            v2f a = *(const v2f*)(aRow + kk);
            const int kr = kk + khalf;
            v2f b0 = { sKV[kr * QPITCH + nA], sKV[(kr + 1) * QPITCH + nA] };
            v2f b1 = { sKV[kr * QPITCH + nB], sKV[(kr + 1) * QPITCH + nB] };
            o0 = __builtin_amdgcn_wmma_f32_16x16x4_f32(false, a, false, b0,
                                                       (short)0, o0, false, false);
            o1 = __builtin_amdgcn_wmma_f32_16x16x4_f32(false, a, false, b1,
                                                       (short)0, o1, false, false);
        }
    }
#pragma unroll
    for (int r = 0; r < 8; ++r) {
        const int m = m0 + msub * 16 + r + hi8;
        Og[((size_t)bh * S + m) * D + nA] = o0[r];
        Og[((size_t)bh * S + m) * D + nB] = o1[r];
    }

#if HAVE_ASYNC_LDS
    wait_async_zero();   // drain p_attn stores (S_ENDPGM would also wait-idle)
#endif
}

extern "C" void kernel_launch(void* const* d_in, const int* in_sizes, int n_in,
                              void* d_out, int out_size, void* d_ws, size_t ws_size,
                              hipStream_t stream) {
    (void)in_sizes; (void)n_in; (void)out_size; (void)d_ws; (void)ws_size;
    const float* Q = (const float*)d_in[0];
    const float* K = (const float*)d_in[1];
    const float* V = (const float*)d_in[2];
    const int*   M = (const int*)d_in[3];
    float* out = (float*)d_out;                       // [96,1024,64]
    float* p   = out + (size_t)BH * S * D;            // [96,1024,1024]

    dim3 grid(S / BM, BH);
    const size_t shmem = (size_t)SMEM_FLOATS * sizeof(float);   // ~303 KB < 320 KB
    attn_gim_fwd<<<grid, 256, shmem, stream>>>(Q, K, V, M, out, p);
}